// CombinedGCN_83459804495955
// MI455X (gfx1250) — compile-verified
//
#include <hip/hip_runtime.h>
#include <cstddef>

typedef float v2f __attribute__((ext_vector_type(2)));
typedef float v8f __attribute__((ext_vector_type(8)));

#define NCOL   128
#define KCHUNK 64
#define BSUB   288   // floats per k-pair sub-row (256 data + 32 pad -> bank-free b64 reads)

// ---------------------------------------------------------------------------
// CSR construction: histogram -> exclusive scan -> fill
// ---------------------------------------------------------------------------
__global__ __launch_bounds__(256) void k_zero_int2(int* a, int* b, int n) {
  int i = blockIdx.x * 256 + threadIdx.x;
  if (i < n) { a[i] = 0; b[i] = 0; }
}

__global__ __launch_bounds__(256) void k_hist(const int* __restrict__ dst,
                                              int* cnt, int e) {
  int i = blockIdx.x * 256 + threadIdx.x;
  if (i < e) atomicAdd(&cnt[dst[i]], 1);
}

__global__ __launch_bounds__(256) void k_scan_block(const int* __restrict__ cnt,
                                                    int* __restrict__ rowStart,
                                                    int* __restrict__ blockSums, int n) {
  __shared__ int sh[256];
  int tid = threadIdx.x;
  int i = blockIdx.x * 256 + tid;
  int x = (i < n) ? cnt[i] : 0;
  sh[tid] = x;
  __syncthreads();
  #pragma unroll
  for (int off = 1; off < 256; off <<= 1) {
    int v = (tid >= off) ? sh[tid - off] : 0;
    __syncthreads();
    sh[tid] += v;
    __syncthreads();
  }
  if (i < n) rowStart[i] = sh[tid] - x;          // exclusive
  if (tid == 255) blockSums[blockIdx.x] = sh[255];
}

__global__ __launch_bounds__(1024) void k_scan_sums(int* bs, int nb) {
  __shared__ int sh[1024];
  int tid = threadIdx.x;
  int x = (tid < nb) ? bs[tid] : 0;
  sh[tid] = x;
  __syncthreads();
  #pragma unroll
  for (int off = 1; off < 1024; off <<= 1) {
    int v = (tid >= off) ? sh[tid - off] : 0;
    __syncthreads();
    sh[tid] += v;
    __syncthreads();
  }
  if (tid < nb) bs[tid] = sh[tid] - x;           // exclusive
}

__global__ __launch_bounds__(256) void k_scan_add(int* rowStart, const int* bs,
                                                  int n, int e) {
  int i = blockIdx.x * 256 + threadIdx.x;
  if (i < n) rowStart[i] += bs[i >> 8];
  if (i == 0) rowStart[n] = e;
}

__global__ __launch_bounds__(256) void k_fill(const int* __restrict__ src,
                                              const int* __restrict__ dst,
                                              const int* __restrict__ rowStart,
                                              int* cursor, int* csrSrc, int e) {
  int i = blockIdx.x * 256 + threadIdx.x;
  if (i >= e) return;
  int d = dst[i];
  int pos = rowStart[d] + atomicAdd(&cursor[d], 1);
  csrSrc[pos] = src[i];
}

__global__ __launch_bounds__(256) void k_dis(const int* __restrict__ cnt,
                                             float* dis, int n) {
  int i = blockIdx.x * 256 + threadIdx.x;
  if (i < n) dis[i] = rsqrtf((float)cnt[i] + 1.0f);   // deg incl. self-loop
}

// ---------------------------------------------------------------------------
// Generic f32 WMMA GEMM:  C[M,128] = [A1 | A2] @ W[K1+K2,128] (+bias)(,relu)
// One 16-row tile per wave. W chunk staged in LDS in *fragment order* so each
// B fragment is a single bank-conflict-free ds_load_b64. A fragments are
// register double-buffered across K-chunks to hide global-load latency
// behind 128 WMMAs.
// ---------------------------------------------------------------------------
__global__ __launch_bounds__(256) void gcn_gemm_wmma(
    const float* __restrict__ A1, int K1,
    const float* __restrict__ A2, int K2,
    const float* __restrict__ W,
    const float* __restrict__ bias,
    float* __restrict__ C, int M, int doRelu)
{
  __shared__ float Bs[32 * BSUB];                 // 36,864 B
  const int K    = K1 + K2;
  const int lane = threadIdx.x & 31;
  const int wave = threadIdx.x >> 5;
  const int hi   = lane >> 4;
  const int ln   = lane & 15;

  const int row0   = (blockIdx.x * 8 + wave) * 16;
  const bool valid = (row0 < M);
  int mrow = row0 + ln;
  if (mrow >= M) mrow = M - 1;                    // clamp loads; stores guarded

  v8f acc[8];
  #pragma unroll
  for (int j = 0; j < 8; ++j) {
    float bv = bias ? bias[j * 16 + ln] : 0.0f;
    v8f z = {bv, bv, bv, bv, bv, bv, bv, bv};
    acc[j] = z;
  }

  auto aBase = [&](int kc) -> const float* {      // lane's A pointer for a chunk
    return (kc < K1) ? (A1 + (size_t)mrow * K1 + kc + 2 * hi)
                     : (A2 + (size_t)mrow * K2 + (kc - K1) + 2 * hi);
  };

  v2f aCur[16], aNxt[16];
  {
    const float* p = aBase(0);
    #pragma unroll
    for (int t = 0; t < 16; ++t) aCur[t] = *(const v2f*)(p + 4 * t);
  }

  for (int kc = 0; kc < K; kc += KCHUNK) {
    __syncthreads();                              // WAR vs previous chunk reads
    // Stage W[kc+2kp+p][n] -> Bs[kp*BSUB + 2n + p]  (fragment-order interleave)
    for (int it = threadIdx.x; it < 32 * 32; it += 256) {
      int kp = it >> 5;
      int c4 = (it & 31) << 2;
      const float* wr = W + (size_t)(kc + 2 * kp) * NCOL + c4;
      float4 r0 = *(const float4*)wr;
      float4 r1 = *(const float4*)(wr + NCOL);
      float4 s0 = {r0.x, r1.x, r0.y, r1.y};
      float4 s1 = {r0.z, r1.z, r0.w, r1.w};
      float* dp = &Bs[kp * BSUB + (c4 << 1)];
      *(float4*)dp       = s0;
      *(float4*)(dp + 4) = s1;
    }
    // Prefetch next chunk's A fragments (waits land after this chunk's WMMAs)
    const bool more = (kc + KCHUNK) < K;
    if (more) {
      const float* p = aBase(kc + KCHUNK);
      #pragma unroll
      for (int t = 0; t < 16; ++t) aNxt[t] = *(const v2f*)(p + 4 * t);
    }
    __syncthreads();

    #pragma unroll
    for (int t = 0; t < 16; ++t) {
      const int sub = (2 * t + hi) * BSUB + 2 * ln;
      #pragma unroll
      for (int j = 0; j < 8; ++j) {
        v2f b = *(const v2f*)&Bs[sub + 32 * j];   // one ds_load_b64
        acc[j] = __builtin_amdgcn_wmma_f32_16x16x4_f32(
            false, aCur[t], false, b, (short)0, acc[j], false, false);
      }
    }
    if (more) {
      #pragma unroll
      for (int t = 0; t < 16; ++t) aCur[t] = aNxt[t];
    }
  }

  if (valid) {
    #pragma unroll
    for (int j = 0; j < 8; ++j) {
      #pragma unroll
      for (int r = 0; r < 8; ++r) {               // VGPR r: M=r / M=r+8 halves
        int rr = row0 + r + 8 * hi;
        if (rr < M) {
          float v = acc[j][r];
          if (doRelu) v = fmaxf(v, 0.0f);
          C[(size_t)rr * NCOL + j * 16 + ln] = v;
        }
      }
    }
  }
}

// ---------------------------------------------------------------------------
// Fused GCN aggregation (gather form, zero float atomics):
//   h[d] = relu( xl[d]*dis[d]^2 + sum_{s in N(d)} xl[s]*dis[s]*dis[d] + bias )
// One wave per destination node; lane holds a float4 of the 128 features.
// ---------------------------------------------------------------------------
__global__ __launch_bounds__(256) void k_aggregate(
    const float* __restrict__ xl, const float* __restrict__ dis,
    const int* __restrict__ rowStart, const int* __restrict__ csrSrc,
    const float* __restrict__ bias, float* __restrict__ hout, int n)
{
  int d    = (blockIdx.x * 256 + threadIdx.x) >> 5;
  int lane = threadIdx.x & 31;
  if (d >= n) return;
  float dd = dis[d];
  float4 a = ((const float4*)(xl + (size_t)d * NCOL))[lane];
  float ss = dd * dd;
  float4 acc;
  acc.x = a.x * ss; acc.y = a.y * ss; acc.z = a.z * ss; acc.w = a.w * ss;
  int beg = rowStart[d], end = rowStart[d + 1];
  for (int i = beg; i < end; ++i) {
    int s = csrSrc[i];                            // wave-uniform broadcast load
    float nrm = dis[s] * dd;
    float4 v = ((const float4*)(xl + (size_t)s * NCOL))[lane];  // 512B coalesced
    acc.x += v.x * nrm; acc.y += v.y * nrm;
    acc.z += v.z * nrm; acc.w += v.w * nrm;
  }
  float4 b4 = ((const float4*)bias)[lane];
  acc.x = fmaxf(acc.x + b4.x, 0.0f);
  acc.y = fmaxf(acc.y + b4.y, 0.0f);
  acc.z = fmaxf(acc.z + b4.z, 0.0f);
  acc.w = fmaxf(acc.w + b4.w, 0.0f);
  ((float4*)(hout + (size_t)d * NCOL))[lane] = acc;
}

// ---------------------------------------------------------------------------
// Head: out[n,2] = log_softmax(h[n,:] @ W_lin + b_lin). One wave per node.
// ---------------------------------------------------------------------------
__global__ __launch_bounds__(256) void k_head(const float* __restrict__ h,
                                              const float* __restrict__ Wl,
                                              const float* __restrict__ bl,
                                              float* __restrict__ out, int m) {
  int wid  = (blockIdx.x * 256 + threadIdx.x) >> 5;
  int lane = threadIdx.x & 31;
  if (wid >= m) return;
  float4 v = ((const float4*)(h + (size_t)wid * NCOL))[lane];
  const float2* W2 = (const float2*)Wl;           // [128][2] row-major
  float2 w0 = W2[lane * 4 + 0], w1 = W2[lane * 4 + 1];
  float2 w2 = W2[lane * 4 + 2], w3 = W2[lane * 4 + 3];
  float s0 = v.x * w0.x + v.y * w1.x + v.z * w2.x + v.w * w3.x;
  float s1 = v.x * w0.y + v.y * w1.y + v.z * w2.y + v.w * w3.y;
  #pragma unroll
  for (int off = 16; off > 0; off >>= 1) {        // wave32 reduction
    s0 += __shfl_down(s0, off, 32);
    s1 += __shfl_down(s1, off, 32);
  }
  if (lane == 0) {
    s0 += bl[0]; s1 += bl[1];
    float mx  = fmaxf(s0, s1);
    float lse = mx + logf(expf(s0 - mx) + expf(s1 - mx));
    out[(size_t)wid * 2 + 0] = s0 - lse;
    out[(size_t)wid * 2 + 1] = s1 - lse;
  }
}

// ---------------------------------------------------------------------------
extern "C" void kernel_launch(void* const* d_in, const int* in_sizes, int n_in,
                              void* d_out, int out_size, void* d_ws, size_t ws_size,
                              hipStream_t stream) {
  const float* high  = (const float*)d_in[0];   // [N,384]
  const float* low   = (const float*)d_in[1];   // [N,64]
  const int*   eidx  = (const int*)d_in[2];     // [2,E]
  const float* W_emb = (const float*)d_in[3];   // [64,128]
  const float* b_emb = (const float*)d_in[4];
  const float* W1    = (const float*)d_in[5];   // [512,128]
  const float* b1    = (const float*)d_in[6];
  const float* W2    = (const float*)d_in[7];   // [128,128]
  const float* b2    = (const float*)d_in[8];
  const float* W_lin = (const float*)d_in[9];   // [128,2]
  const float* b_lin = (const float*)d_in[10];
  float* out = (float*)d_out;

  const int N = in_sizes[0] / 384;
  const int E = in_sizes[2] / 2;
  const int* srcA = eidx;
  const int* dstA = eidx + E;

  // workspace layout (4-byte element offsets)
  float* wsf = (float*)d_ws;
  int*   wsi = (int*)d_ws;
  const size_t PADN  = 131072;                  // >= N slots
  float* dis       = wsf;                       // N floats
  int*   cnt       = wsi + PADN;                // N
  int*   cursor    = wsi + 2 * PADN;            // N
  int*   rowStart  = wsi + 3 * PADN;            // N+1
  int*   blockSums = wsi + 4 * PADN;            // <= 8192
  int*   csrSrc    = wsi + 4 * PADN + 8192;     // E
  size_t oBuf = (4 * PADN + 8192 + (size_t)E + 1023) & ~(size_t)1023;
  const size_t NF = (size_t)N * NCOL;
  float* bufEmb = wsf + oBuf;                   // 51.2 MB each
  float* bufXl  = bufEmb + NF;
  float* bufAgg = bufXl + NF;

  const int nB  = (N + 255) / 256;
  const int eB  = (E + 255) / 256;
  const int nb  = (N + 255) / 256;              // scan blocks (391 <= 1024)
  const int nWav = ((size_t)N * 32 + 255) / 256;
  const int tiles = (N + 15) / 16;
  dim3 gemmGrid((tiles + 7) / 8);

  // --- CSR build + normalization ---
  k_zero_int2 <<<nB, 256, 0, stream>>>(cnt, cursor, N);
  k_hist      <<<eB, 256, 0, stream>>>(dstA, cnt, E);
  k_scan_block<<<nb, 256, 0, stream>>>(cnt, rowStart, blockSums, N);
  k_scan_sums <<<1, 1024, 0, stream>>>(blockSums, nb);
  k_scan_add  <<<nb, 256, 0, stream>>>(rowStart, blockSums, N, E);
  k_fill      <<<eB, 256, 0, stream>>>(srcA, dstA, rowStart, cursor, csrSrc, E);
  k_dis       <<<nB, 256, 0, stream>>>(cnt, dis, N);

  // --- low_emb = relu(low @ W_emb + b_emb) ---
  gcn_gemm_wmma<<<gemmGrid, 256, 0, stream>>>(low, 64, nullptr, 0,
                                              W_emb, b_emb, bufEmb, N, 1);
  // --- xl1 = [high | low_emb] @ W1 ; conv1 aggregate + b1 + relu ---
  gcn_gemm_wmma<<<gemmGrid, 256, 0, stream>>>(high, 384, bufEmb, 128,
                                              W1, nullptr, bufXl, N, 0);
  k_aggregate<<<nWav, 256, 0, stream>>>(bufXl, dis, rowStart, csrSrc,
                                        b1, bufAgg, N);
  // --- xl2 = h1 @ W2 ; conv2 aggregate + b2 + relu ---
  gcn_gemm_wmma<<<gemmGrid, 256, 0, stream>>>(bufAgg, 128, nullptr, 0,
                                              W2, nullptr, bufEmb, N, 0);
  k_aggregate<<<nWav, 256, 0, stream>>>(bufEmb, dis, rowStart, csrSrc,
                                        b2, bufXl, N);
  // --- head + log_softmax ---
  k_head<<<nWav, 256, 0, stream>>>(bufXl, W_lin, b_lin, out, N);
}